// GraphConvolution_38422777430738
// MI455X (gfx1250) — compile-verified
//
#include <hip/hip_runtime.h>

typedef __attribute__((ext_vector_type(2))) float v2f;
typedef __attribute__((ext_vector_type(4))) float v4f;
typedef __attribute__((ext_vector_type(8))) float v8f;

#define D 128  // feature dim == units (fixed by reference)

// ---------------------------------------------------------------------------
// Kernel 1: xw = x @ w  via V_WMMA_F32_16X16X4_F32
// Block = 256 threads = 8 waves. Each wave owns a 16-column slice of W;
// block covers all 128 output columns for one 16-row tile of x per step.
// Persistent grid: each block loops over M-tiles. W (128x128 f32 = 64 KB)
// staged once per block into LDS.
// ---------------------------------------------------------------------------
__global__ __launch_bounds__(256) void gcn_gemm_wmma(
    const float* __restrict__ x, const float* __restrict__ w,
    float* __restrict__ xw, int n_nodes, int mtiles)
{
    __shared__ float lw[D * D];  // W staged, row-major [k][n], 64 KB

    const int tid = threadIdx.x;
    // Cooperative coalesced load of W into LDS (float4, 16 iters/thread)
    for (int i = tid * 4; i < D * D; i += 256 * 4) {
        *(v4f*)(&lw[i]) = *(const v4f*)(w + i);
    }
    __syncthreads();

    const int wave  = tid >> 5;        // 0..7 -> 16-column slice
    const int lane  = tid & 31;
    const int n0    = wave * 16;
    const int mrow  = lane & 15;       // A row within tile
    const int khalf = (lane >> 4) * 2; // A/B K sub-offset per half-wave
    const int ncol  = n0 + (lane & 15);

    for (int t = blockIdx.x; t < mtiles; t += gridDim.x) {
        const int m0 = t * 16;

        v8f acc = {};
        #pragma unroll
        for (int kc = 0; kc < D; kc += 4) {
            const int kb = kc + khalf;
            // A: 16x4 f32 tile; lane L holds x[m0 + (L&15)][kb], [kb+1]
            int arow = m0 + mrow;
            arow = arow < n_nodes ? arow : 0;
            v2f a = *(const v2f*)(x + (size_t)arow * D + kb);
            // B: 4x16 f32 tile from LDS; VGPR0 = W[kb][n], VGPR1 = W[kb+1][n]
            v2f b;
            b.x = lw[kb * D + ncol];
            b.y = lw[(kb + 1) * D + ncol];
            acc = __builtin_amdgcn_wmma_f32_16x16x4_f32(
                false, a, false, b, (short)0, acc, false, false);
        }

        // C/D layout: VGPR r -> row m0 + r + 8*(lane>>4), col n0 + (lane&15)
        const int srow = m0 + (lane >> 4) * 8;
        #pragma unroll
        for (int r = 0; r < 8; ++r) {
            const int row = srow + r;
            if (row < n_nodes) xw[(size_t)row * D + ncol] = acc[r];
        }
    }
}

// ---------------------------------------------------------------------------
// Kernel 2: edge gather + scatter-add.
// One wave (32 lanes) per edge: lane handles 4 consecutive features.
// Gather is a coalesced b128 per lane (512 B / wave from L2-resident xw);
// scatter uses hardware GLOBAL_ATOMIC_ADD_F32 (no CAS loop).
// ---------------------------------------------------------------------------
__global__ __launch_bounds__(256) void gcn_edge_scatter(
    const int* __restrict__ erow, const int* __restrict__ ecol,
    const float* __restrict__ eval, const float* __restrict__ xw,
    float* __restrict__ out, int n_edges)
{
    const int gid  = blockIdx.x * 256 + threadIdx.x;
    const int e    = gid >> 5;
    const int lane = gid & 31;
    if (e >= n_edges) return;

    const int   row = erow[e];
    const int   col = ecol[e];
    const float val = eval[e];

    const int c = lane * 4;
    v4f m = *(const v4f*)(xw + (size_t)col * D + c);
    m *= val;

    float* p = out + (size_t)row * D + c;
    asm volatile(
        "global_atomic_add_f32 %0, %1, off\n\t"
        "global_atomic_add_f32 %0, %2, off offset:4\n\t"
        "global_atomic_add_f32 %0, %3, off offset:8\n\t"
        "global_atomic_add_f32 %0, %4, off offset:12"
        :
        : "v"(p), "v"(m.x), "v"(m.y), "v"(m.z), "v"(m.w)
        : "memory");
}

// ---------------------------------------------------------------------------
// Kernel 3: in-place ReLU, float4 vectorized.
// ---------------------------------------------------------------------------
__global__ __launch_bounds__(256) void gcn_relu(float* __restrict__ out, int n4)
{
    const int i = blockIdx.x * 256 + threadIdx.x;
    if (i >= n4) return;
    v4f v = *(v4f*)(out + (size_t)i * 4);
    v.x = v.x > 0.f ? v.x : 0.f;
    v.y = v.y > 0.f ? v.y : 0.f;
    v.z = v.z > 0.f ? v.z : 0.f;
    v.w = v.w > 0.f ? v.w : 0.f;
    *(v4f*)(out + (size_t)i * 4) = v;
}

extern "C" void kernel_launch(void* const* d_in, const int* in_sizes, int n_in,
                              void* d_out, int out_size, void* d_ws, size_t ws_size,
                              hipStream_t stream)
{
    const float* x    = (const float*)d_in[0];
    const float* w    = (const float*)d_in[1];
    const int*   erow = (const int*)d_in[2];
    const int*   ecol = (const int*)d_in[3];
    const float* eval = (const float*)d_in[4];

    const int n_nodes = in_sizes[0] / D;
    const int n_edges = in_sizes[2];

    float* xw  = (float*)d_ws;   // [n_nodes, 128] f32 scratch (51.2 MB)
    float* out = (float*)d_out;

    // out must start at zero for the atomic scatter-add (graph-capture safe)
    hipMemsetAsync(d_out, 0, (size_t)out_size * sizeof(float), stream);

    // 1) dense projection via f32 WMMA (persistent blocks)
    const int mtiles = (n_nodes + 15) / 16;
    const int gblocks = mtiles < 2048 ? mtiles : 2048;
    gcn_gemm_wmma<<<gblocks, 256, 0, stream>>>(x, w, xw, n_nodes, mtiles);

    // 2) edge gather + atomic scatter (one wave per edge)
    const long long ethreads = (long long)n_edges * 32;
    const int eblocks = (int)((ethreads + 255) / 256);
    gcn_edge_scatter<<<eblocks, 256, 0, stream>>>(erow, ecol, eval, xw, out, n_edges);

    // 3) ReLU
    const int n4 = out_size / 4;
    gcn_relu<<<(n4 + 255) / 256, 256, 0, stream>>>(out, n4);
}